// ViST_GCN_9947144258104
// MI455X (gfx1250) — compile-verified
//
#include <hip/hip_runtime.h>
#include <cmath>

typedef __attribute__((ext_vector_type(2))) float v2f;
typedef __attribute__((ext_vector_type(8))) float v8f;

#define FEAT 128

// ---------------------------------------------------------------- utilities
__global__ __launch_bounds__(256) void zero_kernel(float* __restrict__ p, int n4) {
  int i = blockIdx.x * 256 + threadIdx.x;
  if (i < n4) ((float4*)p)[i] = make_float4(0.f, 0.f, 0.f, 0.f);
}

__global__ __launch_bounds__(256) void degree_kernel(const int* __restrict__ row,
                                                     float* __restrict__ deg,
                                                     int n_edges) {
  int e = blockIdx.x * 256 + threadIdx.x;
  if (e < n_edges) atomicAdd(&deg[row[e]], 1.0f);
}

__global__ __launch_bounds__(256) void recip_kernel(float* __restrict__ deg, int n) {
  int i = blockIdx.x * 256 + threadIdx.x;
  if (i < n) {
    float d = deg[i];
    deg[i] = d > 0.0f ? 1.0f / d : 0.0f;
  }
}

// --------------------------------------------------------------- WMMA GEMM
// Y[M,128] = X[M,128] @ W^T + b      (W is [128 out, 128 in], row-major)
// Block = 256 threads = 8 waves; block tile = 16 rows x 128 cols;
// wave w owns output cols [16w, 16w+16). M must be a multiple of 16
// (100000 = 6250*16) so EXEC is all-1s for every WMMA.
__global__ __launch_bounds__(256) void gemm_bias_wmma(const float* __restrict__ X,
                                                      const float* __restrict__ W,
                                                      const float* __restrict__ bias,
                                                      float* __restrict__ Y) {
  __shared__ float sX[16 * FEAT];  // 8 KB X tile, shared by all 8 waves

  const int tid  = threadIdx.x;
  const int wave = tid >> 5;
  const int lane = tid & 31;
  const int row0 = blockIdx.x * 16;

  // cooperative stage of the 16x128 X tile
  const float4* Xv  = (const float4*)(X + (size_t)row0 * FEAT);
  float4*       sXv = (float4*)sX;
  for (int i = tid; i < 16 * FEAT / 4; i += 256) sXv[i] = Xv[i];
  __syncthreads();

  const int n0 = wave * 16;
  const int mh = lane >> 4;   // lane half: selects K offset 0 or 2
  const int ml = lane & 15;   // M (for A) / N (for B) index within tile

  // A frag per ISA: lanes 0-15 hold (M=ml, K=k..k+1), lanes 16-31 (M=ml, K=k+2..k+3)
  const float* Xm = sX + ml * FEAT + mh * 2;
  // B frag: B[k][n] = W[n][k]; symmetric layout, lane half selects K offset
  const float* Wn = W + (size_t)(n0 + ml) * FEAT + mh * 2;

  v8f c = {};
#pragma unroll 8
  for (int k = 0; k < FEAT; k += 4) {
    v2f a = *(const v2f*)(Xm + k);
    v2f b = *(const v2f*)(Wn + k);
    // (neg_a, A, neg_b, B, c_mod, C, reuse_a, reuse_b)
    c = __builtin_amdgcn_wmma_f32_16x16x4_f32(false, a, false, b, (short)0, c,
                                              false, false);
  }

  // D layout: VGPR r, lanes 0-15 -> M = r ; lanes 16-31 -> M = r + 8 ; N = ml
  const float bn = bias[n0 + ml];
  float* y = Y + (size_t)(row0 + mh * 8) * FEAT + n0 + ml;
#pragma unroll
  for (int r = 0; r < 8; ++r) y[(size_t)r * FEAT] = c[r] + bn;
}

// ------------------------------------------------------------------- SpMM
// One wave per edge: lane loads float4 of H[col], atomically adds into AGG[row].
__global__ __launch_bounds__(256) void spmm_scatter(const float* __restrict__ H,
                                                    const int* __restrict__ row,
                                                    const int* __restrict__ col,
                                                    float* __restrict__ AGG,
                                                    int n_edges) {
  int e = blockIdx.x * 8 + (threadIdx.x >> 5);
  if (e >= n_edges) return;
  int lane = threadIdx.x & 31;
  int r = row[e];
  int c = col[e];
  float4 v = *(const float4*)(H + (size_t)c * FEAT + lane * 4);
  float* dst = AGG + (size_t)r * FEAT + lane * 4;
  atomicAdd(dst + 0, v.x);
  atomicAdd(dst + 1, v.y);
  atomicAdd(dst + 2, v.z);
  atomicAdd(dst + 3, v.w);
}

// scale by 1/deg (mean), optionally ELU, in place
__global__ __launch_bounds__(256) void scale_kernel(float* __restrict__ H,
                                                    const float* __restrict__ deginv,
                                                    int n_elems, int do_elu) {
  int i = blockIdx.x * 256 + threadIdx.x;
  if (i >= n_elems) return;
  int node = i >> 7;  // FEAT == 128
  float h = H[i] * deginv[node];
  if (do_elu) h = h > 0.0f ? h : expf(h) - 1.0f;
  H[i] = h;
}

// ------------------------------------------------------------------ driver
extern "C" void kernel_launch(void* const* d_in, const int* in_sizes, int n_in,
                              void* d_out, int out_size, void* d_ws, size_t ws_size,
                              hipStream_t stream) {
  const float* x          = (const float*)d_in[0];
  const int*   edge_index = (const int*)d_in[1];
  const float* W1         = (const float*)d_in[2];
  const float* b1         = (const float*)d_in[3];
  const float* W2         = (const float*)d_in[4];
  const float* b2         = (const float*)d_in[5];
  float*       out        = (float*)d_out;

  const int n_nodes = in_sizes[0] / FEAT;   // 100000
  const int n_edges = in_sizes[1] / 2;      // 1600000
  const int* row = edge_index;
  const int* col = edge_index + n_edges;

  // workspace: deg | bufA (n_nodes*128) | bufB (n_nodes*128)
  const size_t degBytes = ((size_t)n_nodes * sizeof(float) + 511) & ~(size_t)511;
  float* deg  = (float*)d_ws;
  float* bufA = (float*)((char*)d_ws + degBytes);
  float* bufB = bufA + (size_t)n_nodes * FEAT;

  const int feat_elems = n_nodes * FEAT;
  const int feat_f4    = feat_elems / 4;
  const int deg_f4     = n_nodes / 4;       // 100000 % 4 == 0

  // degree = segment_sum(1); then reciprocal (0 where deg==0)
  zero_kernel<<<(deg_f4 + 255) / 256, 256, 0, stream>>>(deg, deg_f4);
  degree_kernel<<<(n_edges + 255) / 256, 256, 0, stream>>>(row, deg, n_edges);
  recip_kernel<<<(n_nodes + 255) / 256, 256, 0, stream>>>(deg, n_nodes);

  // layer 1: linear -> mean-aggregate -> ELU
  gemm_bias_wmma<<<n_nodes / 16, 256, 0, stream>>>(x, W1, b1, bufA);
  zero_kernel<<<(feat_f4 + 255) / 256, 256, 0, stream>>>(bufB, feat_f4);
  spmm_scatter<<<(n_edges + 7) / 8, 256, 0, stream>>>(bufA, row, col, bufB, n_edges);
  scale_kernel<<<(feat_elems + 255) / 256, 256, 0, stream>>>(bufB, deg, feat_elems, 1);

  // layer 2: linear -> mean-aggregate
  gemm_bias_wmma<<<n_nodes / 16, 256, 0, stream>>>(bufB, W2, b2, bufA);
  zero_kernel<<<(feat_f4 + 255) / 256, 256, 0, stream>>>(out, feat_f4);
  spmm_scatter<<<(n_edges + 7) / 8, 256, 0, stream>>>(bufA, row, col, out, n_edges);
  scale_kernel<<<(feat_elems + 255) / 256, 256, 0, stream>>>(out, deg, feat_elems, 0);
}